// JacTorch_13408887898249
// MI455X (gfx1250) — compile-verified
//
#include <hip/hip_runtime.h>

// Jacobi relaxation, 5-point Laplacian, N=512 grid, batch B=16, 20 sweeps.
// Bandwidth-bound (0.5 GFLOP vs ~1 GB moved; 48 MB working set is L2-resident
// on MI455X's 192 MB L2). Uses the CDNA5 async global->LDS copy path to stage
// halo tiles so each x value is read from L2 ~once and reused 4x from LDS.

#define GN     512
#define GB     16
#define TILE   32
#define HALO_S (TILE + 2)                     // 34
#define HALO_N (HALO_S * HALO_S)              // 1156
#define TAIL_N (HALO_N - 4 * 256)             // 132
#define ITERS  20

#ifndef __has_builtin
#define __has_builtin(x) 0
#endif

typedef __attribute__((address_space(1))) int as1_int;   // global ("__device__")
typedef __attribute__((address_space(3))) int as3_int;   // LDS ("__shared__")

#if __has_builtin(__builtin_amdgcn_global_load_async_to_lds_b32)
// Signature (per hipcc diagnostic): (AS1 int* gsrc, AS3 int* ldst, imm, imm cpol)
#define ASYNC_LOAD_B32(dst_lds, src_glob)                                     \
    __builtin_amdgcn_global_load_async_to_lds_b32(                            \
        (as1_int*)(src_glob), (as3_int*)(dst_lds), 0, 0)
#else
#define ASYNC_LOAD_B32(dst_lds, src_glob)                                     \
    asm volatile("global_load_async_to_lds_b32 %0, %1, off"                   \
                 :: "v"((unsigned)(unsigned long long)(dst_lds)),             \
                    "v"(src_glob)                                             \
                 : "memory")
#endif

#if __has_builtin(__builtin_amdgcn_s_wait_asynccnt)
#define WAIT_ASYNC0() __builtin_amdgcn_s_wait_asynccnt(0)
#else
#define WAIT_ASYNC0() asm volatile("s_wait_asynccnt 0x0" ::: "memory")
#endif

__global__ __launch_bounds__(256)
void jacobi_step_kernel(const float* __restrict__ xin,
                        const float* __restrict__ bvec,
                        const float* __restrict__ invD,
                        float* __restrict__ xout)
{
    __shared__ float sx[HALO_N];   // 34x34 staging tile (~4.6 KB of 320 KB LDS)

    const int tileC = blockIdx.x * TILE;
    const int tileR = blockIdx.y * TILE;
    const int batch = blockIdx.z;

    const float* __restrict__ xb = xin  + (size_t)batch * GN * GN;
    const float* __restrict__ bb = bvec + (size_t)batch * GN * GN;
    float*       __restrict__ ob = xout + (size_t)batch * GN * GN;

    const int tid = threadIdx.y * 16 + threadIdx.x;   // 0..255

    // Halo element (lr,lc) for staging round k is flat index tid + 256k;
    // track row/col incrementally: 256 = 7*34 + 18 (one conditional carry).
    int lr = tid / HALO_S;
    int lc = tid - lr * HALO_S;

    // Block-uniform: does the whole 34x34 halo window lie inside the grid?
    const bool interior = (tileR >= 1) && (tileR + TILE + 1 <= GN) &&
                          (tileC >= 1) && (tileC + TILE + 1 <= GN);

    if (interior) {
        // Fast path (196/256 tiles): no zero-fill, no extra barrier,
        // unconditional straight-line async global->LDS loads (ASYNCcnt).
        const float* src0 = xb + (size_t)(tileR - 1) * GN + (tileC - 1);
        #pragma unroll
        for (int k = 0; k < 5; ++k) {
            if (k < 4 || tid < TAIL_N)
                ASYNC_LOAD_B32(&sx[lr * HALO_S + lc], src0 + lr * GN + lc);
            lr += 7; lc += 18;
            if (lc >= HALO_S) { lc -= HALO_S; lr += 1; }
        }
    } else {
        // Boundary tiles: out-of-domain halo entries must contribute 0
        // (clamped stencil), so pre-zero then load only in-domain entries.
        for (int i = tid; i < HALO_N; i += 256)
            sx[i] = 0.0f;
        __syncthreads();
        #pragma unroll
        for (int k = 0; k < 5; ++k) {
            const int gr = tileR - 1 + lr;
            const int gc = tileC - 1 + lc;
            if ((k < 4 || tid < TAIL_N) &&
                (unsigned)gr < (unsigned)GN && (unsigned)gc < (unsigned)GN)
                ASYNC_LOAD_B32(&sx[lr * HALO_S + lc], xb + (size_t)gr * GN + gc);
            lr += 7; lc += 18;
            if (lc >= HALO_S) { lc -= HALO_S; lr += 1; }
        }
    }
    WAIT_ASYNC0();       // per-wave: all async LDS writes landed
    __syncthreads();     // workgroup-wide visibility

    // Each thread produces 4 points at (ty+16a, tx+16c) in the tile:
    // coalesced 64B segments for b / invD loads and the store.
    const int tx = threadIdx.x, ty = threadIdx.y;
    #pragma unroll
    for (int a = 0; a < 2; ++a) {
        #pragma unroll
        for (int c2 = 0; c2 < 2; ++c2) {
            const int r = ty + 16 * a;        // 0..31 tile-local row
            const int c = tx + 16 * c2;       // 0..31 tile-local col
            const int gi = (tileR + r) * GN + (tileC + c);
            // M_vals are all -1 => b - Mx = b + neighbor sum.
            const float nsum = sx[(r    ) * HALO_S + (c + 1)]    // up
                             + sx[(r + 2) * HALO_S + (c + 1)]    // down
                             + sx[(r + 1) * HALO_S + (c    )]    // left
                             + sx[(r + 1) * HALO_S + (c + 2)];   // right
            ob[gi] = (bb[gi] + nsum) * invD[gi];
        }
    }
}

extern "C" void kernel_launch(void* const* d_in, const int* in_sizes, int n_in,
                              void* d_out, int out_size, void* d_ws, size_t ws_size,
                              hipStream_t stream)
{
    (void)in_sizes; (void)n_in; (void)out_size; (void)ws_size;

    const float* u    = (const float*)d_in[0];  // (B,1,N,N) == (B, N*N) contiguous
    const float* bvec = (const float*)d_in[1];  // (B, N*N)
    // d_in[2..4] = M_rows/M_cols/M_vals: the fixed 5-point stencil with all
    // off-diagonal values -1 -> folded into the kernel's neighbor sum.
    const float* invD = (const float*)d_in[5];  // (N*N,)
    // d_in[6] = maxiter (== 20, compile-time constant of the reference; device
    // scalars can't be read on host during graph capture).

    float* out = (float*)d_out;
    float* ws  = (float*)d_ws;   // needs GB*GN*GN floats = 16 MiB scratch

    dim3 grid(GN / TILE, GN / TILE, GB);   // (16,16,16) = 4096 workgroups
    dim3 block(16, 16, 1);                 // 8 wave32 waves

    // Ping-pong so sweep #20 lands in d_out:
    // it=0: u->ws, it=1: ws->out, it=2: out->ws, ..., it=19: ws->out.
    for (int it = 0; it < ITERS; ++it) {
        const float* src = (it == 0) ? u : ((it & 1) ? ws : out);
        float*       dst = (it & 1) ? out : ws;
        jacobi_step_kernel<<<grid, block, 0, stream>>>(src, bvec, invD, dst);
    }
}